// LFSR_44805098832260
// MI455X (gfx1250) — compile-verified
//
#include <hip/hip_runtime.h>

typedef __attribute__((ext_vector_type(2))) float v2f;
typedef __attribute__((ext_vector_type(8))) float v8f;

#define N_NODES 169
#define M_PAD   176          // 11 * 16, zero-padded M for WMMA tiles
#define FEAT    1024
#define H_IMG   169
#define W_IMG   1024
#define HW      (H_IMG * W_IMG)     // 173056 = 16 * 10816
#define N_EDGES 5408
#define K2      576                  // 64 channels * 9 taps

// CDNA5 async global->LDS copy path (guarded: falls back to plain LDS stores)
#if defined(__has_builtin)
#if __has_builtin(__builtin_amdgcn_global_load_async_to_lds_b32) && \
    __has_builtin(__builtin_amdgcn_s_wait_asynccnt)
#define HAVE_ASYNC_LDS 1
#endif
#endif
#ifndef HAVE_ASYNC_LDS
#define HAVE_ASYNC_LDS 0
#endif

#if HAVE_ASYNC_LDS
typedef __attribute__((address_space(1))) int as1_int;   // global (device)
typedef __attribute__((address_space(3))) int as3_int;   // LDS
#endif

// ---------------------------------------------------------------------------
// Zero the (padded) segment-sum accumulator. 176*1024 floats, float4 stores.
__global__ __launch_bounds__(256) void k_zero(float* __restrict__ p) {
    int i = (blockIdx.x * 256 + threadIdx.x) * 4;
    *(float4*)(p + i) = make_float4(0.f, 0.f, 0.f, 0.f);
}

// ---------------------------------------------------------------------------
// GCN message passing: h0[dst[e]] += input[src[e]]  (one block per edge)
__global__ __launch_bounds__(256) void k_segsum(const float* __restrict__ in,
                                                const int* __restrict__ src,
                                                const int* __restrict__ dst,
                                                float* __restrict__ h0) {
    int e = blockIdx.x;
    int s = src[e];
    int d = dst[e];
    const float* srow = in + (size_t)s * FEAT;
    float*       drow = h0 + (size_t)d * FEAT;
    for (int i = threadIdx.x; i < FEAT; i += 256)
        atomicAdd(&drow[i], srow[i]);
}

// ---------------------------------------------------------------------------
// Repack W2 [32][64][3][3] -> Wp [32][9 taps][64 ci]  (tap-major K ordering)
__global__ __launch_bounds__(256) void k_repack_w2(const float* __restrict__ W2,
                                                   float* __restrict__ Wp) {
    int i = blockIdx.x * 256 + threadIdx.x;
    if (i >= 32 * K2) return;
    int co = i / K2, k = i - co * K2;
    int ci = k / 9,  tap = k - ci * 9;
    Wp[co * K2 + tap * 64 + ci] = W2[i];
}

// ---------------------------------------------------------------------------
// h1 = h0 @ W_lin^T + b_lin   via V_WMMA_F32_16X16X4_F32
// One wave per 16x16 output tile. 11 M-tiles x 64 N-tiles = 704 tiles.
__global__ __launch_bounds__(256) void k_linear_wmma(const float* __restrict__ h0,
                                                     const float* __restrict__ W,
                                                     const float* __restrict__ bias,
                                                     float* __restrict__ h1) {
    int wave = threadIdx.x >> 5;
    int lane = threadIdx.x & 31;
    int tile = blockIdx.x * 8 + wave;       // 0..703
    int mt = tile % 11;
    int nt = tile / 11;
    int m0 = mt * 16, n0 = nt * 16;
    int hl  = lane >> 4;                    // 0: K offsets {0,1}, 1: {2,3}
    int r16 = lane & 15;

    const float* arow = h0 + (size_t)(m0 + r16) * FEAT;  // A row (M index)
    const float* brow = W  + (size_t)(n0 + r16) * FEAT;  // B col (N index), W[n,k]
    __builtin_prefetch(arow, 0, 0);
    __builtin_prefetch(brow, 0, 0);

    v8f acc = {};
    for (int k = 0; k < FEAT; k += 4) {
        v2f a = *(const v2f*)(arow + k + hl * 2);
        v2f b = *(const v2f*)(brow + k + hl * 2);
        acc = __builtin_amdgcn_wmma_f32_16x16x4_f32(
            false, a, false, b, (short)0, acc, false, false);
    }

    int   n  = n0 + r16;
    float bv = bias[n];
    int   mb = m0 + hl * 8;
#pragma unroll
    for (int r = 0; r < 8; ++r) {
        int m = mb + r;
        if (m < N_NODES) h1[(size_t)m * FEAT + n] = acc[r] + bv;
    }
}

// ---------------------------------------------------------------------------
// conv1: 1 -> 64 channels, 3x3, SAME, relu.  One thread per pixel;
// the 9 input taps live in registers and are reused across all 64 channels.
__global__ __launch_bounds__(256) void k_conv1(const float* __restrict__ h1,
                                               const float* __restrict__ W1,
                                               const float* __restrict__ b1,
                                               float* __restrict__ c1) {
    __shared__ float w[64 * 9];
    __shared__ float bs[64];
    for (int i = threadIdx.x; i < 64 * 9; i += 256) w[i] = W1[i];
    if (threadIdx.x < 64) bs[threadIdx.x] = b1[threadIdx.x];
    __syncthreads();

    int p = blockIdx.x * 256 + threadIdx.x;
    if (p >= HW) return;
    int y = p >> 10, x = p & 1023;

    float v[9];
#pragma unroll
    for (int t = 0; t < 9; ++t) {
        int yy = y + t / 3 - 1, xx = x + t % 3 - 1;
        v[t] = (yy >= 0 && yy < H_IMG && xx >= 0 && xx < W_IMG)
                   ? h1[yy * W_IMG + xx] : 0.f;
    }
    for (int ch = 0; ch < 64; ++ch) {
        float s = bs[ch];
#pragma unroll
        for (int t = 0; t < 9; ++t) s = fmaf(w[ch * 9 + t], v[t], s);
        c1[(size_t)ch * HW + p] = fmaxf(s, 0.f);
    }
}

// ---------------------------------------------------------------------------
// conv2 (dominant): 64 -> 32 channels, 3x3, relu, as implicit GEMM with an
// LDS-staged halo tile.  Block = 64-pixel row segment; 8 waves =
// 4 M-subtiles x 2 N-tiles.  K ordered tap-major: k = tap*64 + ci, so the
// inner loop is 2 ds_loads + 1 v2f global load + 1 wmma, no div/mod.
__global__ __launch_bounds__(256) void k_conv2_wmma(const float* __restrict__ c1,
                                                    const float* __restrict__ Wp,
                                                    const float* __restrict__ b2,
                                                    float* __restrict__ c2) {
    __shared__ float At[64 * 3 * 72];       // [ci][dy][72], 66 columns used
    int tid  = threadIdx.x;
    int sblk = blockIdx.x;                  // 169*16 = 2704 supertiles
    int ty = sblk >> 4;                     // image row
    int tx = (sblk & 15) * 64;              // x0 of 64-pixel segment

    // --- stage A halo tile (zero-padded) into LDS -------------------------
    for (int idx = tid; idx < 64 * 3 * 66; idx += 256) {
        int xo = idx % 66;                  // 0..65  -> x = tx + xo - 1
        int t2 = idx / 66;                  // ci*3 + dy
        int dy = t2 % 3;
        int ci = t2 / 3;
        int yy = ty + dy - 1;
        int xx = tx + xo - 1;
        int l  = t2 * 72 + xo;
        if (yy >= 0 && yy < H_IMG && xx >= 0 && xx < W_IMG) {
#if HAVE_ASYNC_LDS
            __builtin_amdgcn_global_load_async_to_lds_b32(
                (as1_int*)(c1 + (size_t)ci * HW + yy * W_IMG + xx),
                (as3_int*)(At + l), 0, 0);
#else
            At[l] = c1[(size_t)ci * HW + yy * W_IMG + xx];
#endif
        } else {
            At[l] = 0.f;
        }
    }
#if HAVE_ASYNC_LDS
    __builtin_amdgcn_s_wait_asynccnt(0);
#endif
    __syncthreads();

    // --- implicit GEMM over the staged tile -------------------------------
    int wave = tid >> 5, lane = tid & 31;
    int nt   = wave & 1;                    // N-tile (output-channel group)
    int msub = wave >> 1;                   // M-subtile within segment
    int hl   = lane >> 4, r16 = lane & 15;
    int xl   = msub * 16 + r16;             // A-row local pixel 0..63
    int co   = nt * 16 + r16;               // B-col output channel
    const float* brow = Wp + (size_t)co * K2;
    __builtin_prefetch(brow, 0, 0);

    v8f acc = {};
    for (int k = 0; k < K2; k += 4) {
        int tap   = k >> 6;                 // wave-uniform (SALU)
        int dy    = tap / 3;
        int dx    = tap - dy * 3;
        int cbase = (k & 63) + hl * 2;      // ci for a.x; a.y is ci+1
        int base  = dy * 72 + xl + dx;
        v2f a, b;
        a.x = At[cbase * 216 + base];
        a.y = At[cbase * 216 + base + 216];
        b   = *(const v2f*)(brow + k + hl * 2);
        acc = __builtin_amdgcn_wmma_f32_16x16x4_f32(
            false, a, false, b, (short)0, acc, false, false);
    }

    float bv = b2[co];
    int   pix0 = ty * W_IMG + tx + msub * 16 + hl * 8;
#pragma unroll
    for (int r = 0; r < 8; ++r)
        c2[(size_t)co * HW + pix0 + r] = fmaxf(acc[r] + bv, 0.f);
}

// ---------------------------------------------------------------------------
// conv3: 32 -> 1 channel, 3x3, relu. One thread per pixel.
__global__ __launch_bounds__(256) void k_conv3(const float* __restrict__ c2,
                                               const float* __restrict__ W3,
                                               const float* __restrict__ b3,
                                               float* __restrict__ out) {
    __shared__ float w[32 * 9];
    for (int i = threadIdx.x; i < 32 * 9; i += 256) w[i] = W3[i];
    __syncthreads();

    int p = blockIdx.x * 256 + threadIdx.x;
    if (p >= HW) return;
    int y = p >> 10, x = p & 1023;

    float s = b3[0];
    for (int ci = 0; ci < 32; ++ci) {
        const float* base = c2 + (size_t)ci * HW;
#pragma unroll
        for (int t = 0; t < 9; ++t) {
            int yy = y + t / 3 - 1, xx = x + t % 3 - 1;
            if (yy >= 0 && yy < H_IMG && xx >= 0 && xx < W_IMG)
                s = fmaf(w[ci * 9 + t], base[yy * W_IMG + xx], s);
        }
    }
    out[p] = fmaxf(s, 0.f);
}

// ---------------------------------------------------------------------------
extern "C" void kernel_launch(void* const* d_in, const int* in_sizes, int n_in,
                              void* d_out, int out_size, void* d_ws, size_t ws_size,
                              hipStream_t stream) {
    const float* input = (const float*)d_in[0];
    const int*   src   = (const int*)  d_in[1];
    const int*   dst   = (const int*)  d_in[2];
    const float* W_lin = (const float*)d_in[3];
    const float* b_lin = (const float*)d_in[4];
    const float* W1    = (const float*)d_in[5];
    const float* b1    = (const float*)d_in[6];
    const float* W2    = (const float*)d_in[7];
    const float* b2    = (const float*)d_in[8];
    const float* W3    = (const float*)d_in[9];
    const float* b3    = (const float*)d_in[10];
    float* out = (float*)d_out;

    // Workspace layout (fp32):
    float* h0 = (float*)d_ws;            // [176][1024] padded segment-sum
    float* h1 = h0 + M_PAD * FEAT;       // [176][1024] after linear
    float* c1 = h1 + M_PAD * FEAT;       // [64][169][1024]
    float* c2 = c1 + (size_t)64 * HW;    // [32][169][1024]
    float* Wp = c2 + (size_t)32 * HW;    // [32][576] tap-major repacked W2

    k_repack_w2 <<<72,             256, 0, stream>>>(W2, Wp);
    k_zero      <<<176,            256, 0, stream>>>(h0);
    k_segsum    <<<N_EDGES,        256, 0, stream>>>(input, src, dst, h0);
    k_linear_wmma<<<88,            256, 0, stream>>>(h0, W_lin, b_lin, h1);
    k_conv1     <<<(HW + 255)/256, 256, 0, stream>>>(h1, W1, b1, c1);
    k_conv2_wmma<<<2704,           256, 0, stream>>>(c1, Wp, b2, c2);
    k_conv3     <<<(HW + 255)/256, 256, 0, stream>>>(c2, W3, b3, out);
}